// QuantumGenerator_52939766890525
// MI455X (gfx1250) — compile-verified
//
#include <hip/hip_runtime.h>
#include <math.h>

#define N_QUBITS 12
#define DEPTH    4
#define HID      64
#define NPARAMS  144      // DEPTH * N_QUBITS * 3
#define BATCH    4096
#define DIM      4096     // 2^N_QUBITS
#define OUT_DIM  2

typedef __attribute__((ext_vector_type(2))) float v2f;
typedef __attribute__((ext_vector_type(8))) float v8f;

// ---------------------------------------------------------------------------
// fp32 WMMA 16x16x4 (CDNA5):  D = A(16x4) * B(4x16) + C(16x16)
// ---------------------------------------------------------------------------
__device__ __forceinline__ v8f wmma_f32(v2f a, v2f b, v8f c) {
  return __builtin_amdgcn_wmma_f32_16x16x4_f32(
      /*neg_a=*/false, a, /*neg_b=*/false, b,
      /*c_mod=*/(short)0, c, /*reuse_a=*/false, /*reuse_b=*/false);
}

// A fragment: rows row0..row0+15, cols k0..k0+3 of row-major src (ld = lda).
// lanes 0-15: M=lane, {K=k0,k0+1}; lanes 16-31: M=lane-16, {K=k0+2,k0+3}
__device__ __forceinline__ v2f load_a_frag(const float* src, int lda,
                                           int row0, int k0, int lane) {
  int m = row0 + (lane & 15);
  int k = k0 + ((lane >> 4) << 1);
  v2f a;
  a.x = src[m * lda + k];
  a.y = src[m * lda + k + 1];
  return a;
}

// B fragment: rows k0..k0+3, cols n0..n0+15 of row-major W (ld = ldb).
__device__ __forceinline__ v2f load_b_frag(const float* W, int ldb,
                                           int k0, int n0, int lane) {
  int c = n0 + (lane & 15);
  int k = k0 + ((lane >> 4) << 1);
  v2f b;
  b.x = W[k * ldb + c];
  b.y = W[(k + 1) * ldb + c];
  return b;
}

// ---------------------------------------------------------------------------
// Kernel 1: circ_params = tanh(noise@W1+b1) @ W2 + b2      [4096,144]
// 4 waves/block, 16 rows per wave, 64 rows/block, 64 blocks.
// ---------------------------------------------------------------------------
__global__ __launch_bounds__(128) void front_mlp_kernel(
    const float* __restrict__ noise, const float* __restrict__ W1,
    const float* __restrict__ b1,    const float* __restrict__ W2,
    const float* __restrict__ b2,    float* __restrict__ params) {
  __shared__ float hbuf[4][16 * HID];          // 16 KB
  const int lane  = threadIdx.x & 31;
  const int wave  = threadIdx.x >> 5;
  const int row0  = (blockIdx.x * 4 + wave) * 16;
  const int col   = lane & 15;
  const int rhalf = (lane >> 4) * 8;
  float* h = hbuf[wave];

  // layer 1: [16,12] @ [12,64] -> tanh -> LDS
  for (int nt = 0; nt < HID / 16; ++nt) {
    v8f c = {};
#pragma unroll
    for (int kc = 0; kc < N_QUBITS; kc += 4) {
      v2f a = load_a_frag(noise, N_QUBITS, row0, kc, lane);
      v2f b = load_b_frag(W1, HID, kc, nt * 16, lane);
      c = wmma_f32(a, b, c);
    }
    int n = nt * 16 + col;
    float bias = b1[n];
#pragma unroll
    for (int r = 0; r < 8; ++r)
      h[(rhalf + r) * HID + n] = tanhf(c[r] + bias);
  }
  __syncthreads();

  // layer 2: [16,64] @ [64,144] + b2 -> params
  for (int nt = 0; nt < NPARAMS / 16; ++nt) {
    v8f c = {};
#pragma unroll
    for (int kc = 0; kc < HID; kc += 4) {
      v2f a = load_a_frag(h, HID, 0, kc, lane);
      v2f b = load_b_frag(W2, NPARAMS, kc, nt * 16, lane);
      c = wmma_f32(a, b, c);
    }
    int n = nt * 16 + col;
    float bias = b2[n];
#pragma unroll
    for (int r = 0; r < 8; ++r)
      params[(size_t)(row0 + rhalf + r) * NPARAMS + n] = c[r] + bias;
  }
}

// ---------------------------------------------------------------------------
// Kernel 2: batched 12-qubit state-vector sim, one block per batch element.
// State lives in LDS (2 ping-pong complex buffers = 64 KB). CNOT ring is
// collapsed into a single permutation pass per layer.
// ---------------------------------------------------------------------------
__global__ __launch_bounds__(256) void quantum_kernel(
    const float* __restrict__ params, float* __restrict__ meas) {
  __shared__ float s_re0[DIM], s_im0[DIM], s_re1[DIM], s_im1[DIM];  // 64 KB
  const int tid = threadIdx.x;
  const int bidx = blockIdx.x;

  for (int i = tid; i < DIM; i += 256) { s_re0[i] = 0.0f; s_im0[i] = 0.0f; }
  if (tid == 0) s_re0[0] = 1.0f;
  __syncthreads();

  const float* p = params + (size_t)bidx * NPARAMS;   // uniform -> scalar loads

  for (int l = 0; l < DEPTH; ++l) {
    float* Ar = (l & 1) ? s_re1 : s_re0;
    float* Ai = (l & 1) ? s_im1 : s_im0;
    float* Br = (l & 1) ? s_re0 : s_re1;
    float* Bi = (l & 1) ? s_im0 : s_im1;

    // 12 fused single-qubit U3 = RZ(c)RY(b)RX(a) gates
    for (int q = 0; q < N_QUBITS; ++q) {
      const float aa = p[(l * N_QUBITS + q) * 3 + 0];
      const float bb = p[(l * N_QUBITS + q) * 3 + 1];
      const float cc = p[(l * N_QUBITS + q) * 3 + 2];
      float sa, ca, sb, cb, sz, cz;
      sincosf(aa * 0.5f, &sa, &ca);
      sincosf(bb * 0.5f, &sb, &cb);
      sincosf(cc * 0.5f, &sz, &cz);
      // RY*RX entries (complex), then RZ phase rows
      const float m00r =  cb * ca, m00i =  sb * sa;
      const float m01r = -sb * ca, m01i = -cb * sa;
      const float m10r =  sb * ca, m10i = -cb * sa;
      const float m11r =  cb * ca, m11i = -sb * sa;
      const float U00r = cz * m00r + sz * m00i, U00i = cz * m00i - sz * m00r;
      const float U01r = cz * m01r + sz * m01i, U01i = cz * m01i - sz * m01r;
      const float U10r = cz * m10r - sz * m10i, U10i = cz * m10i + sz * m10r;
      const float U11r = cz * m11r - sz * m11i, U11i = cz * m11i + sz * m11r;

      const int bitpos  = N_QUBITS - 1 - q;   // wire 0 = MSB
      const int stride  = 1 << bitpos;
      const int lowmask = stride - 1;
      for (int t = tid; t < DIM / 2; t += 256) {
        int i0 = ((t & ~lowmask) << 1) | (t & lowmask);
        int i1 = i0 | stride;
        float xr = Ar[i0], xi = Ai[i0];
        float yr = Ar[i1], yi = Ai[i1];
        Ar[i0] = U00r * xr - U00i * xi + U01r * yr - U01i * yi;
        Ai[i0] = U00r * xi + U00i * xr + U01r * yi + U01i * yr;
        Ar[i1] = U10r * xr - U10i * xi + U11r * yr - U11i * yi;
        Ai[i1] = U10r * xi + U10i * xr + U11r * yi + U11i * yr;
      }
      __syncthreads();
    }

    // Ring of CNOTs (q -> (q+1)%12) as one basis permutation, scatter A -> B
    for (int i = tid; i < DIM; i += 256) {
      int x = i;
#pragma unroll
      for (int q = 0; q < N_QUBITS; ++q) {
        int pc = N_QUBITS - 1 - q;
        int pt = N_QUBITS - 1 - ((q + 1) % N_QUBITS);
        if ((x >> pc) & 1) x ^= (1 << pt);
      }
      Br[x] = Ar[i];
      Bi[x] = Ai[i];
    }
    __syncthreads();
  }

  // final state is in buffer 0 (4 permutes: 0->1->0->1->0). PauliZ expvals.
  float acc[N_QUBITS];
#pragma unroll
  for (int q = 0; q < N_QUBITS; ++q) acc[q] = 0.0f;
  for (int i = tid; i < DIM; i += 256) {
    float pr = s_re0[i] * s_re0[i] + s_im0[i] * s_im0[i];
#pragma unroll
    for (int q = 0; q < N_QUBITS; ++q)
      acc[q] += ((i >> (N_QUBITS - 1 - q)) & 1) ? -pr : pr;
  }
  float* part = s_re1;  // 256*12 floats scratch (buffer 1 is free now)
#pragma unroll
  for (int q = 0; q < N_QUBITS; ++q) part[tid * N_QUBITS + q] = acc[q];
  __syncthreads();
  for (int s = 128; s > 0; s >>= 1) {
    if (tid < s) {
#pragma unroll
      for (int q = 0; q < N_QUBITS; ++q)
        part[tid * N_QUBITS + q] += part[(tid + s) * N_QUBITS + q];
    }
    __syncthreads();
  }
  if (tid < N_QUBITS) meas[(size_t)bidx * N_QUBITS + tid] = part[tid];
}

// ---------------------------------------------------------------------------
// Kernel 3: out = tanh(meas@W3+b3) @ W4 + b4         [4096,2]
// ---------------------------------------------------------------------------
__global__ __launch_bounds__(128) void back_mlp_kernel(
    const float* __restrict__ meas, const float* __restrict__ W3,
    const float* __restrict__ b3,   const float* __restrict__ W4,
    const float* __restrict__ b4,   float* __restrict__ out) {
  __shared__ float hbuf[4][16 * HID];
  const int lane  = threadIdx.x & 31;
  const int wave  = threadIdx.x >> 5;
  const int row0  = (blockIdx.x * 4 + wave) * 16;
  const int col   = lane & 15;
  const int rhalf = (lane >> 4) * 8;
  float* h = hbuf[wave];

  for (int nt = 0; nt < HID / 16; ++nt) {
    v8f c = {};
#pragma unroll
    for (int kc = 0; kc < N_QUBITS; kc += 4) {
      v2f a = load_a_frag(meas, N_QUBITS, row0, kc, lane);
      v2f b = load_b_frag(W3, HID, kc, nt * 16, lane);
      c = wmma_f32(a, b, c);
    }
    int n = nt * 16 + col;
    float bias = b3[n];
#pragma unroll
    for (int r = 0; r < 8; ++r)
      h[(rhalf + r) * HID + n] = tanhf(c[r] + bias);
  }
  __syncthreads();

  // final layer: N=2, one zero-padded 16-wide WMMA tile
  v8f c = {};
#pragma unroll
  for (int kc = 0; kc < HID; kc += 4) {
    v2f a = load_a_frag(h, HID, 0, kc, lane);
    int k = kc + ((lane >> 4) << 1);
    v2f b;
    b.x = (col < OUT_DIM) ? W4[k * OUT_DIM + col] : 0.0f;
    b.y = (col < OUT_DIM) ? W4[(k + 1) * OUT_DIM + col] : 0.0f;
    c = wmma_f32(a, b, c);
  }
  if (col < OUT_DIM) {
    float bias = b4[col];
#pragma unroll
    for (int r = 0; r < 8; ++r)
      out[(size_t)(row0 + rhalf + r) * OUT_DIM + col] = c[r] + bias;
  }
}

// ---------------------------------------------------------------------------
extern "C" void kernel_launch(void* const* d_in, const int* in_sizes, int n_in,
                              void* d_out, int out_size, void* d_ws, size_t ws_size,
                              hipStream_t stream) {
  const float* noise = (const float*)d_in[0];
  const float* W1 = (const float*)d_in[1];
  const float* b1 = (const float*)d_in[2];
  const float* W2 = (const float*)d_in[3];
  const float* b2 = (const float*)d_in[4];
  const float* W3 = (const float*)d_in[5];
  const float* b3 = (const float*)d_in[6];
  const float* W4 = (const float*)d_in[7];
  const float* b4 = (const float*)d_in[8];
  float* out = (float*)d_out;

  float* params = (float*)d_ws;                          // [4096,144]
  float* meas   = params + (size_t)BATCH * NPARAMS;      // [4096,12]

  front_mlp_kernel<<<BATCH / 64, 128, 0, stream>>>(noise, W1, b1, W2, b2, params);
  quantum_kernel<<<BATCH, 256, 0, stream>>>(params, meas);
  back_mlp_kernel<<<BATCH / 64, 128, 0, stream>>>(meas, W3, b3, W4, b4, out);
}